// BiLinearLayer_72670846648710
// MI455X (gfx1250) — compile-verified
//
#include <hip/hip_runtime.h>

// Problem constants (match reference)
constexpr int NF     = 40;      // fields
constexpr int NBATCH = 16384;   // batch
constexpr int ND     = 32;      // feature dim
constexpr int NOUT   = 50;      // output dim
constexpr int NPAIRS = NF * (NF - 1) / 2;  // 780

// Tiling
constexpr int NB  = 16;   // batches per workgroup
constexpr int SB  = 8;    // batches per sub-tile (one wave per batch in phase B)
constexpr int FT  = 48;   // NF padded to 3 WMMA tiles
constexpr int LDF = 33;   // LDS row stride for sF/sV (odd -> conflict-free row gathers)
constexpr int PK  = 784;  // NPAIRS padded to multiple of 16 (phase-C K)
constexpr int LDP = 785;  // LDS row stride for sp (odd)

typedef __attribute__((ext_vector_type(2))) float v2f;
typedef __attribute__((ext_vector_type(8))) float v8f;

__device__ __forceinline__ int pair_index(int f, int g) {
    // order of jnp.triu_indices(F, k=1)
    return f * NF - (f * (f + 1)) / 2 + (g - f - 1);
}

__global__ __launch_bounds__(256) void bilinear_fused_kernel(
    const float* __restrict__ fields, const float* __restrict__ W,
    const float* __restrict__ bw, const float* __restrict__ Wc,
    const float* __restrict__ bc, float* __restrict__ out)
{
    __shared__ float sF[SB][FT][LDF];   // fields tile  (b, f, d)
    __shared__ float sV[SB][FT][LDF];   // v_dot tile   (b, f, e)
    __shared__ float sp[NB][LDP];       // pair values  (b, pair)
    __shared__ float sW[ND][ND + 1];
    __shared__ float sCscr[4][16][17];  // phase-C cross-wave reduction scratch
    __shared__ float sbw[ND];
    __shared__ float sbc[64];

    const int tid  = threadIdx.x;
    const int wave = tid >> 5;
    const int lane = tid & 31;
    const int hi   = lane >> 4;   // lane half: selects K offset {0,1} vs {2,3}
    const int lo   = lane & 15;
    const int batch0 = blockIdx.x * NB;

    // ---- Phase 0: stage W / bw / bc, zero sp pad columns and sF pad rows ----
    for (int i = tid; i < ND * ND; i += 256) sW[i / ND][i % ND] = W[i];
    if (tid < ND) sbw[tid] = bw[tid];
    if (tid < 64) sbc[tid] = (tid < NOUT) ? bc[tid] : 0.0f;
    for (int i = tid; i < NB * (LDP - NPAIRS); i += 256) {
        int r = i / (LDP - NPAIRS);
        int c = NPAIRS + i % (LDP - NPAIRS);
        sp[r][c] = 0.0f;
    }
    for (int i = tid; i < SB * (FT - NF) * LDF; i += 256) {
        int b   = i / ((FT - NF) * LDF);
        int rem = i % ((FT - NF) * LDF);
        sF[b][NF + rem / LDF][rem % LDF] = 0.0f;   // pad rows, written once
    }
    __syncthreads();

    for (int s = 0; s < NB / SB; ++s) {
        const int gb = batch0 + s * SB;

        // ---- Async stage fields tile straight into LDS (bypass VGPRs) ----
        // 16B chunk per lane: 8 chunks per (b,f) row of 32 floats.
        for (int c = tid; c < SB * NF * 8; c += 256) {
            const int b    = c / (NF * 8);
            const int rem  = c % (NF * 8);
            const int f    = rem >> 3;
            const int part = rem & 7;
            unsigned ldsoff = (unsigned)(size_t)&sF[b][f][part * 4];
            const float* g  = &fields[((size_t)f * NBATCH + (gb + b)) * ND + part * 4];
            asm volatile("global_load_async_to_lds_b128 %0, %1, off"
                         :: "v"(ldsoff), "v"(g) : "memory");
        }
        asm volatile("s_wait_asynccnt 0x0" ::: "memory");
        __syncthreads();

        // ---- Phase A: v_dot = fields @ W + bw (24 tile jobs, 3 per wave) ----
        for (int j = 0; j < 3; ++j) {
            const int job = wave * 3 + j;        // 0..23
            const int b  = job / 3;
            const int ti = job % 3;
            v8f acc0, acc1;
            {
                const float bw0 = sbw[lo];
                const float bw1 = sbw[16 + lo];
                #pragma unroll
                for (int r = 0; r < 8; ++r) { acc0[r] = bw0; acc1[r] = bw1; }
            }
            #pragma unroll
            for (int kk = 0; kk < 8; ++kk) {
                const int k0 = kk * 4 + 2 * hi;
                v2f a, b0, b1;
                a.x  = sF[b][ti * 16 + lo][k0];
                a.y  = sF[b][ti * 16 + lo][k0 + 1];
                b0.x = sW[k0][lo];      b0.y = sW[k0 + 1][lo];
                b1.x = sW[k0][16 + lo]; b1.y = sW[k0 + 1][16 + lo];
                acc0 = __builtin_amdgcn_wmma_f32_16x16x4_f32(
                    false, a, false, b0, (short)0, acc0, false, false);
                acc1 = __builtin_amdgcn_wmma_f32_16x16x4_f32(
                    false, a, false, b1, (short)0, acc1, false, false);
            }
            #pragma unroll
            for (int r = 0; r < 8; ++r) {
                sV[b][ti * 16 + r + 8 * hi][lo]      = acc0[r];
                sV[b][ti * 16 + r + 8 * hi][16 + lo] = acc1[r];
            }
        }
        __syncthreads();

        // ---- Phase B: per-batch pairwise = v_dot @ fields^T (upper-tri) ----
        {
            const int b = wave;   // one batch per wave
            for (int ti = 0; ti < 3; ++ti) {
                v2f afr[8];       // hoist A fragments across the tj loop
                #pragma unroll
                for (int kk = 0; kk < 8; ++kk) {
                    const int k0 = kk * 4 + 2 * hi;
                    afr[kk].x = sV[b][ti * 16 + lo][k0];
                    afr[kk].y = sV[b][ti * 16 + lo][k0 + 1];
                }
                for (int tj = ti; tj < 3; ++tj) {
                    v8f acc = {};
                    #pragma unroll
                    for (int kk = 0; kk < 8; ++kk) {
                        const int k0 = kk * 4 + 2 * hi;
                        v2f bf;
                        bf.x = sF[b][tj * 16 + lo][k0];     // B[k][g] = sF[g][k]
                        bf.y = sF[b][tj * 16 + lo][k0 + 1];
                        acc = __builtin_amdgcn_wmma_f32_16x16x4_f32(
                            false, afr[kk], false, bf, (short)0, acc, false, false);
                    }
                    const int g = tj * 16 + lo;
                    #pragma unroll
                    for (int r = 0; r < 8; ++r) {
                        const int f = ti * 16 + r + 8 * hi;
                        if (f < NF && g < NF && f < g)
                            sp[s * SB + b][pair_index(f, g)] = acc[r];
                    }
                }
            }
        }
        __syncthreads();   // protects sF/sV reuse next s, and sp for phase C
    }

    // ---- Phase C: out = sp(16x784) @ Wc(784x64pad) + bc ; all 8 waves ----
    {
        const int tn   = wave & 3;
        const int half = wave >> 2;          // K split across wave pairs
        const int col  = tn * 16 + lo;
        v8f acc;
        {
            const float init = (half == 0) ? sbc[col] : 0.0f;
            #pragma unroll
            for (int r = 0; r < 8; ++r) acc[r] = init;
        }
        for (int kk = 0; kk < PK / 8; ++kk) {        // 98 k-steps per half
            const int k0 = half * (PK / 2) + kk * 4 + 2 * hi;
            v2f a, bf;
            a.x = sp[lo][k0];
            a.y = sp[lo][k0 + 1];
            bf.x = (k0 < NPAIRS && col < NOUT)     ? Wc[(size_t)k0 * NOUT + col]       : 0.0f;
            bf.y = (k0 + 1 < NPAIRS && col < NOUT) ? Wc[(size_t)(k0 + 1) * NOUT + col] : 0.0f;
            acc = __builtin_amdgcn_wmma_f32_16x16x4_f32(
                false, a, false, bf, (short)0, acc, false, false);
        }
        if (half == 1) {
            #pragma unroll
            for (int r = 0; r < 8; ++r) sCscr[tn][r + 8 * hi][lo] = acc[r];
        }
        __syncthreads();
        if (half == 0) {
            #pragma unroll
            for (int r = 0; r < 8; ++r) acc[r] += sCscr[tn][r + 8 * hi][lo];
            if (col < NOUT) {
                #pragma unroll
                for (int r = 0; r < 8; ++r) {
                    const int m = r + 8 * hi;   // local batch
                    out[(size_t)(batch0 + m) * NOUT + col] = acc[r];
                }
            }
        }
    }
}

extern "C" void kernel_launch(void* const* d_in, const int* in_sizes, int n_in,
                              void* d_out, int out_size, void* d_ws, size_t ws_size,
                              hipStream_t stream) {
    const float* fields = (const float*)d_in[0];
    const float* W      = (const float*)d_in[1];
    const float* bw     = (const float*)d_in[2];
    const float* Wc     = (const float*)d_in[3];
    const float* bc     = (const float*)d_in[4];
    float* out = (float*)d_out;

    dim3 grid(NBATCH / NB);   // 1024 workgroups
    dim3 block(256);          // 8 wave32s
    hipLaunchKernelGGL(bilinear_fused_kernel, grid, block, 0, stream,
                       fields, W, bw, Wc, bc, out);
}